// GNNPlusLayer_28372553957731
// MI455X (gfx1250) — compile-verified
//
#include <hip/hip_runtime.h>
#include <hip/hip_bf16.h>

// ---------------------------------------------------------------------------
// GNN layer for MI455X (gfx1250, wave32).
// h_pre = agg@W_nbr + x@W_root + b ; h = relu(BN1(h_pre)) + x
// y_pre = relu(h@w1+b1)@w2 + b2 + h ; out = BN2(y_pre)
// Memory-bound by the 1.6M-edge scatter (~1 GB @ 23.3 TB/s); GEMMs (4.9 GF)
// run in full f32 with V_WMMA_F32_16X16X4_F32 (precision-exact, free).
// Weights are stored K-pair-major so every B-fragment is one aligned 8-byte
// load straight into an even VGPR pair (no repack movs before the WMMA).
// ---------------------------------------------------------------------------

#define N_NODES 100000
#define N_EDGES 1600000
#define D_CH 64
#define H_CH 128
#define ND (N_NODES * D_CH)
#define N_TILES (N_NODES / 16)   // 6250, exact

typedef float v2f __attribute__((ext_vector_type(2)));
typedef float v8f __attribute__((ext_vector_type(8)));

__device__ __forceinline__ v8f wmma_f32(v2f a, v2f b, v8f c) {
    // D(16x16,f32) = A(16x4,f32) * B(4x16,f32) + C
    return __builtin_amdgcn_wmma_f32_16x16x4_f32(false, a, false, b,
                                                 (short)0, c, false, false);
}

__device__ __forceinline__ v8f splat8(float v) {
    return (v8f){v, v, v, v, v, v, v, v};
}

// ---------------------------------------------------------------------------
// 0) pre-pack FFN weights K-pair-major: Wp[(k>>1)*C*2 + col*2 + (k&1)]
// ---------------------------------------------------------------------------
__global__ __launch_bounds__(256) void pack_w_kernel(
    const float* __restrict__ w1, const float* __restrict__ w2,
    float* __restrict__ w1p, float* __restrict__ w2p) {
    unsigned i = blockIdx.x * 256u + threadIdx.x;
    if (i < D_CH * H_CH) {            // w1: 64x128
        unsigned k = i >> 7, col = i & 127u;
        w1p[((k >> 1) * H_CH + col) * 2 + (k & 1u)] = w1[i];
    }
    if (i < H_CH * D_CH) {            // w2: 128x64
        unsigned k = i >> 6, col = i & 63u;
        w2p[((k >> 1) * D_CH + col) * 2 + (k & 1u)] = w2[i];
    }
}

// ---------------------------------------------------------------------------
// 1) scatter-add: agg[dst] += x[src].  One wave per edge, 2 channels per lane.
// ---------------------------------------------------------------------------
__global__ __launch_bounds__(256) void scatter_add_kernel(
    const float* __restrict__ x,
    const long long* __restrict__ src,
    const long long* __restrict__ dst,
    float* __restrict__ agg) {
    unsigned gid = blockIdx.x * 256u + threadIdx.x;
    unsigned e = gid >> 5;
    if (e >= N_EDGES) return;
    unsigned c = (gid & 31u) * 2u;
    long long s = src[e];
    long long d = dst[e];
    float2 v = *reinterpret_cast<const float2*>(x + (size_t)s * D_CH + c);
    atomicAdd(agg + (size_t)d * D_CH + c, v.x);
    atomicAdd(agg + (size_t)d * D_CH + c + 1, v.y);
}

// ---------------------------------------------------------------------------
// 2) GraphConv GEMM: one wave computes a 16x64 tile of
//    h_pre = agg@W_nbr + x@W_root + b, and accumulates BN1 column stats.
//    Weights staged in LDS in K-pair-major layout.
// ---------------------------------------------------------------------------
__global__ __launch_bounds__(256) void gnn_gemm_kernel(
    const float* __restrict__ agg, const float* __restrict__ x,
    const float* __restrict__ Wn, const float* __restrict__ Wr,
    const float* __restrict__ bias,
    float* __restrict__ hpre,
    float* __restrict__ sum, float* __restrict__ sumsq) {
    __shared__ float sWn[D_CH * D_CH];
    __shared__ float sWr[D_CH * D_CH];
    for (int i = threadIdx.x; i < D_CH * D_CH; i += 256) {
        int k = i >> 6, col = i & 63;
        int p = ((k >> 1) * D_CH + col) * 2 + (k & 1);
        sWn[p] = Wn[i];
        sWr[p] = Wr[i];
    }
    __syncthreads();

    unsigned wave = blockIdx.x * 8u + (threadIdx.x >> 5);
    if (wave >= N_TILES) return;          // wave-uniform, EXEC stays all-ones
    unsigned lane = threadIdx.x & 31u;
    unsigned rs = lane & 15u;             // row (A) / column (B,C) select
    unsigned hi = lane >> 4;              // half-wave select
    unsigned m0 = wave * 16u;

    const float* aggRow = agg + (size_t)(m0 + rs) * D_CH + 2u * hi;
    const float* xRow   = x   + (size_t)(m0 + rs) * D_CH + 2u * hi;

    v8f acc[4];
#pragma unroll
    for (int t = 0; t < 4; ++t) acc[t] = splat8(bias[t * 16 + rs]);

#pragma unroll
    for (int kk = 0; kk < D_CH; kk += 4) {
        // A frags: lanes 0-15 hold K={kk,kk+1}, lanes 16-31 K={kk+2,kk+3}
        v2f aA = *reinterpret_cast<const v2f*>(aggRow + kk);
        v2f aX = *reinterpret_cast<const v2f*>(xRow + kk);
        // packed base for K-pair (kk + 2*hi)/2 = kk/2 + hi
        unsigned pb = ((kk >> 1) + hi) * (D_CH * 2);
#pragma unroll
        for (int t = 0; t < 4; ++t) {
            unsigned col = t * 16u + rs;
            v2f bN = *reinterpret_cast<const v2f*>(sWn + pb + col * 2u);
            v2f bR = *reinterpret_cast<const v2f*>(sWr + pb + col * 2u);
            acc[t] = wmma_f32(aA, bN, acc[t]);
            acc[t] = wmma_f32(aX, bR, acc[t]);
        }
    }

    // epilogue: store tile + per-channel partial sums for BN1
#pragma unroll
    for (int t = 0; t < 4; ++t) {
        unsigned col = t * 16u + rs;
        float s = 0.f, s2 = 0.f;
#pragma unroll
        for (int r = 0; r < 8; ++r) {
            float v = acc[t][r];
            hpre[(size_t)(m0 + hi * 8u + r) * D_CH + col] = v;
            s += v;
            s2 += v * v;
        }
        atomicAdd(sum + col, s);
        atomicAdd(sumsq + col, s2);
    }
}

// ---------------------------------------------------------------------------
// 3) BN finalize: per-channel scale/shift from accumulated sums.
// ---------------------------------------------------------------------------
__global__ void bn_finalize_kernel(const float* __restrict__ sum,
                                   const float* __restrict__ sumsq,
                                   const float* __restrict__ gamma,
                                   const float* __restrict__ beta,
                                   float* __restrict__ scale,
                                   float* __restrict__ shift) {
    int c = threadIdx.x;
    if (c >= D_CH) return;
    float inv_n = 1.0f / (float)N_NODES;
    float mean = sum[c] * inv_n;
    float var = sumsq[c] * inv_n - mean * mean;
    float sc = gamma[c] * rsqrtf(var + 1e-5f);
    scale[c] = sc;
    shift[c] = beta[c] - mean * sc;
}

// ---------------------------------------------------------------------------
// 4) h = relu(BN1(h_pre)) + x   (float4 vectorized)
// ---------------------------------------------------------------------------
__global__ __launch_bounds__(256) void bn_relu_res_kernel(
    const float* __restrict__ hpre, const float* __restrict__ x,
    const float* __restrict__ scale, const float* __restrict__ shift,
    float* __restrict__ h) {
    unsigned i = blockIdx.x * 256u + threadIdx.x;
    if (i >= ND / 4) return;
    unsigned c4 = i & 15u;  // 64 channels = 16 float4 per row
    float4 v = reinterpret_cast<const float4*>(hpre)[i];
    float4 xr = reinterpret_cast<const float4*>(x)[i];
    float4 sc = reinterpret_cast<const float4*>(scale)[c4];
    float4 sh = reinterpret_cast<const float4*>(shift)[c4];
    float4 o;
    o.x = fmaxf(v.x * sc.x + sh.x, 0.f) + xr.x;
    o.y = fmaxf(v.y * sc.y + sh.y, 0.f) + xr.y;
    o.z = fmaxf(v.z * sc.z + sh.z, 0.f) + xr.z;
    o.w = fmaxf(v.w * sc.w + sh.w, 0.f) + xr.w;
    reinterpret_cast<float4*>(h)[i] = o;
}

// ---------------------------------------------------------------------------
// 5) FFN: y_pre = relu(h@w1+b1)@w2 + b2 + h, plus BN2 stats.
//    One wave per 16-row tile; C-layout -> A-layout re-stripe via private LDS.
//    B-fragments come from pre-packed global weights (single b64 loads).
// ---------------------------------------------------------------------------
__global__ __launch_bounds__(128) void ffn_kernel(
    const float* __restrict__ h,
    const float* __restrict__ w1p, const float* __restrict__ b1,
    const float* __restrict__ w2p, const float* __restrict__ b2,
    float* __restrict__ ypre,
    float* __restrict__ sum, float* __restrict__ sumsq) {
    __shared__ float sT[4 * 16 * H_CH];   // 32 KB: per-wave 16x128 transpose buf
    unsigned wv = threadIdx.x >> 5;
    unsigned wave = blockIdx.x * 4u + wv;
    if (wave >= N_TILES) return;
    unsigned lane = threadIdx.x & 31u;
    unsigned rs = lane & 15u;
    unsigned hi = lane >> 4;
    unsigned m0 = wave * 16u;

    const float* hRow = h + (size_t)(m0 + rs) * D_CH + 2u * hi;

    // ---- stage 1: t = relu(h @ w1 + b1), 16x128 in 8 column tiles ----
    v8f acc1[8];
#pragma unroll
    for (int t = 0; t < 8; ++t) acc1[t] = splat8(b1[t * 16 + rs]);

#pragma unroll
    for (int kk = 0; kk < D_CH; kk += 4) {
        v2f a = *reinterpret_cast<const v2f*>(hRow + kk);
        unsigned pb = ((kk >> 1) + hi) * (H_CH * 2);
#pragma unroll
        for (int t = 0; t < 8; ++t) {
            unsigned col = t * 16u + rs;
            v2f b = *reinterpret_cast<const v2f*>(w1p + pb + col * 2u);
            acc1[t] = wmma_f32(a, b, acc1[t]);
        }
    }

    // relu + re-stripe through wave-private LDS (DS ops in-order per wave)
    float* tb = sT + wv * 16 * H_CH;
#pragma unroll
    for (int t = 0; t < 8; ++t) {
        unsigned col = t * 16u + rs;
#pragma unroll
        for (int r = 0; r < 8; ++r)
            tb[(hi * 8u + r) * H_CH + col] = fmaxf(acc1[t][r], 0.f);
    }

    // ---- stage 2: y_pre = t @ w2 + b2 + h ----
    v8f acc2[4];
#pragma unroll
    for (int t = 0; t < 4; ++t) acc2[t] = splat8(b2[t * 16 + rs]);

    const float* tRow = tb + rs * H_CH + 2u * hi;
#pragma unroll
    for (int kk = 0; kk < H_CH; kk += 4) {
        v2f a = *reinterpret_cast<const v2f*>(tRow + kk);  // ds_load_b64
        unsigned pb = ((kk >> 1) + hi) * (D_CH * 2);
#pragma unroll
        for (int t = 0; t < 4; ++t) {
            unsigned col = t * 16u + rs;
            v2f b = *reinterpret_cast<const v2f*>(w2p + pb + col * 2u);
            acc2[t] = wmma_f32(a, b, acc2[t]);
        }
    }

    // epilogue: residual, store, BN2 stats
#pragma unroll
    for (int t = 0; t < 4; ++t) {
        unsigned col = t * 16u + rs;
        float s = 0.f, s2 = 0.f;
#pragma unroll
        for (int r = 0; r < 8; ++r) {
            size_t row = (size_t)(m0 + hi * 8u + r);
            float v = acc2[t][r] + h[row * D_CH + col];
            ypre[row * D_CH + col] = v;
            s += v;
            s2 += v * v;
        }
        atomicAdd(sum + col, s);
        atomicAdd(sumsq + col, s2);
    }
}

// ---------------------------------------------------------------------------
// 6) out = BN2(y_pre)
// ---------------------------------------------------------------------------
__global__ __launch_bounds__(256) void bn_apply_kernel(
    const float* __restrict__ ypre,
    const float* __restrict__ scale, const float* __restrict__ shift,
    float* __restrict__ out) {
    unsigned i = blockIdx.x * 256u + threadIdx.x;
    if (i >= ND / 4) return;
    unsigned c4 = i & 15u;
    float4 v = reinterpret_cast<const float4*>(ypre)[i];
    float4 sc = reinterpret_cast<const float4*>(scale)[c4];
    float4 sh = reinterpret_cast<const float4*>(shift)[c4];
    float4 o;
    o.x = v.x * sc.x + sh.x;
    o.y = v.y * sc.y + sh.y;
    o.z = v.z * sc.z + sh.z;
    o.w = v.w * sc.w + sh.w;
    reinterpret_cast<float4*>(out)[i] = o;
}

// ---------------------------------------------------------------------------
extern "C" void kernel_launch(void* const* d_in, const int* in_sizes, int n_in,
                              void* d_out, int out_size, void* d_ws, size_t ws_size,
                              hipStream_t stream) {
    const float* x = (const float*)d_in[0];
    const long long* ei = (const long long*)d_in[1];   // int64 [2, E]
    const float* W_root = (const float*)d_in[2];
    const float* W_nbr = (const float*)d_in[3];
    const float* b_gnn = (const float*)d_in[4];
    const float* bn1_g = (const float*)d_in[5];
    const float* bn1_b = (const float*)d_in[6];
    const float* ffn_w1 = (const float*)d_in[7];
    const float* ffn_b1 = (const float*)d_in[8];
    const float* ffn_w2 = (const float*)d_in[9];
    const float* ffn_b2 = (const float*)d_in[10];
    const float* bn2_g = (const float*)d_in[11];
    const float* bn2_b = (const float*)d_in[12];
    float* out = (float*)d_out;

    const long long* src = ei;
    const long long* dst = ei + N_EDGES;

    // workspace layout (floats)
    float* ws = (float*)d_ws;
    float* agg   = ws;                 // [N, D]
    float* hpre  = ws + (size_t)ND;    // [N, D], reused for y_pre
    float* hbuf  = ws + (size_t)2 * ND;// [N, D]
    float* stats = ws + (size_t)3 * ND;
    float* sum1 = stats, *sumsq1 = stats + 64;
    float* sum2 = stats + 128, *sumsq2 = stats + 192;
    float* scale1 = stats + 256, *shift1 = stats + 320;
    float* scale2 = stats + 384, *shift2 = stats + 448;
    float* w1p = stats + 512;                  // 8192 floats
    float* w2p = w1p + (size_t)D_CH * H_CH;    // 8192 floats
    float* ypre = hpre;  // reuse after hpre consumed

    hipMemsetAsync(agg, 0, (size_t)ND * sizeof(float), stream);
    hipMemsetAsync(stats, 0, 256 * sizeof(float), stream);

    // 0) pre-pack FFN weights (K-pair-major)
    pack_w_kernel<<<(H_CH * D_CH + 255) / 256, 256, 0, stream>>>(
        ffn_w1, ffn_w2, w1p, w2p);
    // 1) scatter-add
    {
        unsigned threads = (unsigned)N_EDGES * 32u;
        scatter_add_kernel<<<threads / 256, 256, 0, stream>>>(x, src, dst, agg);
    }
    // 2) GraphConv GEMM + BN1 stats
    gnn_gemm_kernel<<<(N_TILES + 7) / 8, 256, 0, stream>>>(
        agg, x, W_nbr, W_root, b_gnn, hpre, sum1, sumsq1);
    // 3) BN1 finalize
    bn_finalize_kernel<<<1, 64, 0, stream>>>(sum1, sumsq1, bn1_g, bn1_b, scale1, shift1);
    // 4) h = relu(BN1(h_pre)) + x
    bn_relu_res_kernel<<<(ND / 4 + 255) / 256, 256, 0, stream>>>(
        hpre, x, scale1, shift1, hbuf);
    // 5) FFN + residual + BN2 stats
    ffn_kernel<<<(N_TILES + 3) / 4, 128, 0, stream>>>(
        hbuf, w1p, ffn_b1, w2p, ffn_b2, ypre, sum2, sumsq2);
    // 6) BN2 finalize
    bn_finalize_kernel<<<1, 64, 0, stream>>>(sum2, sumsq2, bn2_g, bn2_b, scale2, shift2);
    // 7) out = BN2(y_pre)
    bn_apply_kernel<<<(ND / 4 + 255) / 256, 256, 0, stream>>>(ypre, scale2, shift2, out);
}